// MulticlassNMS_83958020702835
// MI455X (gfx1250) — compile-verified
//
#include <hip/hip_runtime.h>
#include <stdint.h>

// ---------------------------------------------------------------------------
// MulticlassNMS for MI455X (gfx1250, wave32).
// Latency-bound greedy NMS; no matmul structure -> no WMMA. CDNA5 paths used:
// async global->LDS (ASYNCcnt) + s_wait_asynccnt, wave32 shuffle reductions
// (ds_swizzle/ds_bpermute), global_prefetch_b8.
// ---------------------------------------------------------------------------

#define AS1 __attribute__((address_space(1)))
#define AS3 __attribute__((address_space(3)))

#if defined(__has_builtin)
# if __has_builtin(__builtin_amdgcn_global_load_async_to_lds_b32)
#  define HAVE_ASYNC_B32 1
# endif
# if __has_builtin(__builtin_amdgcn_s_wait_asynccnt)
#  define HAVE_WAIT_ASYNC 1
# endif
#endif

#define NMS_T    512          // 16 wave32 waves per class-block
#define NMS_W    (NMS_T / 32) // 16 waves
#define SLOTS    8            // supports n <= 4096 boxes
#define TOPK_T   1024
#define TOPK_W   (TOPK_T / 32) // 32 waves
#define NEGSENT  (-1.0e30f)

// argmax combine with lax.top_k / stable-argsort tie-break (lower index wins)
__device__ __forceinline__ void amax2(float& v, int& i, float v2, int i2) {
  if (v2 > v || (v2 == v && i2 < i)) { v = v2; i = i2; }
}

// wave32 xor-tournament argmax (no barriers; lowers to DS permute ops)
__device__ __forceinline__ void wave_argmax(float& v, int& i) {
  #pragma unroll
  for (int off = 16; off > 0; off >>= 1) {
    float v2 = __shfl_xor(v, off, 32);
    int   i2 = __shfl_xor(i, off, 32);
    amax2(v, i, v2, i2);
  }
}

// ---------------- Kernel 1: box prep (x1,y1,x2,y2,area as SoA) -------------
__global__ void nms_prep_kernel(const float* __restrict__ bboxes,
                                float* __restrict__ x1, float* __restrict__ y1,
                                float* __restrict__ x2, float* __restrict__ y2,
                                float* __restrict__ area, int n) {
  int i = blockIdx.x * blockDim.x + threadIdx.x;
  if (i < n) {
    float a = bboxes[i * 4 + 0];
    float b = bboxes[i * 4 + 1];
    float c = bboxes[i * 4 + 2];
    float d = bboxes[i * 4 + 3];
    x1[i] = a; y1[i] = b; x2[i] = c; y2[i] = d;
    area[i] = (c - a) * (d - b);
  }
}

// ---------------- Kernel 2: transpose scores[:,1:] -> [class][box] ---------
__global__ void nms_transpose_kernel(const float* __restrict__ scores,
                                     float* __restrict__ scoresT,
                                     int n, int C, int Ctot) {
  int i = blockIdx.x * blockDim.x + threadIdx.x;
  if (i < n * C) {
    int c = i / n;
    int b = i - c * n;
    scoresT[i] = scores[(size_t)b * Ctot + c + 1];  // skip background class 0
  }
}

// ---------------- Kernel 3: per-class greedy NMS ---------------------------
// One block per class. Boxes live in registers; class scores staged via
// async global->LDS. finals[c*n+j] = kept ? score : -1  (may alias scoresT).
__global__ __launch_bounds__(NMS_T)
void nms_class_kernel(const float* __restrict__ x1, const float* __restrict__ y1,
                      const float* __restrict__ x2, const float* __restrict__ y2,
                      const float* __restrict__ area,
                      const float* __restrict__ scoresT,
                      float* __restrict__ finals,
                      const float* __restrict__ conf_p,
                      const float* __restrict__ nms_p, int n) {
  __shared__ float sscore[SLOTS * NMS_T];
  __shared__ float rv[NMS_W];
  __shared__ int   ri[NMS_W];
  __shared__ float best_v_sh;
  __shared__ int   best_i_sh;
  __shared__ float bco[8];

  const int t    = threadIdx.x;
  const int lane = t & 31;
  const int wid  = t >> 5;
  const int c    = blockIdx.x;
  const float confT = conf_p[0];
  const float nmsT  = nms_p[0];
  const float* srow = scoresT + (size_t)c * n;

  // ---- Stage this class's score row into LDS (CDNA5 async copy path) ----
#ifdef HAVE_ASYNC_B32
  for (int i = t; i < n; i += NMS_T) {
    __builtin_amdgcn_global_load_async_to_lds_b32(
        (AS1 int*)(srow + i), (AS3 int*)&sscore[i], 0, 0);
  }
# ifdef HAVE_WAIT_ASYNC
  __builtin_amdgcn_s_wait_asynccnt(0);
# else
  asm volatile("s_wait_asynccnt 0" ::: "memory");
# endif
#else
  for (int i = t; i < n; i += NMS_T) sscore[i] = srow[i];
#endif
  __syncthreads();

  // ---- Load owned boxes into registers (coalesced; prefetch hint) ----
  __builtin_prefetch(&x1[t], 0, 0);
  __builtin_prefetch(&area[t], 0, 0);

  float rx1[SLOTS], ry1[SLOTS], rx2[SLOTS], ry2[SLOTS], rar[SLOTS];
  float rsc[SLOTS], ro[SLOTS];
  unsigned kmask = 0u;
  #pragma unroll
  for (int s = 0; s < SLOTS; ++s) {
    int j = t + s * NMS_T;
    if (j < n) {
      rx1[s] = x1[j]; ry1[s] = y1[j]; rx2[s] = x2[j]; ry2[s] = y2[j];
      rar[s] = area[j];
      float sc = sscore[j];
      ro[s]  = sc;
      rsc[s] = (sc > confT) ? sc : NEGSENT;  // valid = score > conf_thresh
    } else {
      rx1[s] = ry1[s] = rx2[s] = ry2[s] = rar[s] = 0.0f;
      ro[s]  = -1.0f;
      rsc[s] = NEGSENT;
    }
  }

  // ---- Greedy NMS: argmax over remaining, suppress by IoU ----
  for (int it = 0; it < n; ++it) {
    // thread-local argmax (ascending s + strict > => lowest index on ties)
    float bv = NEGSENT; int bi = 0x7fffffff;
    #pragma unroll
    for (int s = 0; s < SLOTS; ++s) {
      if (rsc[s] > bv) { bv = rsc[s]; bi = t + s * NMS_T; }
    }
    // stage 1: intra-wave reduction (barrier-free DS shuffles)
    wave_argmax(bv, bi);
    if (lane == 0) { rv[wid] = bv; ri[wid] = bi; }
    __syncthreads();                               // barrier 1
    // stage 2: wave 0 reduces the 16 per-wave partials
    if (t < 32) {
      float v = (lane < NMS_W) ? rv[lane] : NEGSENT;
      int   i = (lane < NMS_W) ? ri[lane] : 0x7fffffff;
      wave_argmax(v, i);
      if (lane == 0) { best_v_sh = v; best_i_sh = i; }
    }
    __syncthreads();                               // barrier 2
    float bestv = best_v_sh;
    int   sel   = best_i_sh;
    if (bestv == NEGSENT) break;  // uniform across block

    if (t == (sel & (NMS_T - 1))) {          // owner thread broadcasts winner
      int s = sel / NMS_T;
      bco[0] = rx1[s]; bco[1] = ry1[s]; bco[2] = rx2[s]; bco[3] = ry2[s];
      bco[4] = rar[s];
      rsc[s] = NEGSENT;                       // processed
      kmask |= (1u << s);                     // kept
    }
    __syncthreads();                               // barrier 3

    const float X1 = bco[0], Y1 = bco[1], X2 = bco[2], Y2 = bco[3], AR = bco[4];
    #pragma unroll
    for (int s = 0; s < SLOTS; ++s) {
      if (rsc[s] != NEGSENT) {
        float ix1 = fmaxf(X1, rx1[s]);
        float iy1 = fmaxf(Y1, ry1[s]);
        float ix2 = fminf(X2, rx2[s]);
        float iy2 = fminf(Y2, ry2[s]);
        float iw = fmaxf(ix2 - ix1, 0.0f);
        float ih = fmaxf(iy2 - iy1, 0.0f);
        float inter = iw * ih;
        float iou = inter / (AR + rar[s] - inter + 1e-9f);
        if (iou > nmsT) rsc[s] = NEGSENT;     // suppressed
      }
    }
    // next iteration's rv/ri writes are fenced by barrier 3 above
  }

  // ---- Emit final masked scores ----
  #pragma unroll
  for (int s = 0; s < SLOTS; ++s) {
    int j = t + s * NMS_T;
    if (j < n) finals[(size_t)c * n + j] = ((kmask >> s) & 1u) ? ro[s] : -1.0f;
  }
}

// ---------------- Kernel 4: iterative top-m over finals (L2-resident) ------
__global__ __launch_bounds__(TOPK_T)
void nms_topk_kernel(float* __restrict__ finals, int total, int m,
                     float* __restrict__ topv, int* __restrict__ topi) {
  __shared__ float rv[TOPK_W];
  __shared__ int   ri[TOPK_W];
  const int t    = threadIdx.x;
  const int lane = t & 31;
  const int wid  = t >> 5;
  for (int k = 0; k < m; ++k) {
    float bv = NEGSENT; int bi = 0x7fffffff;
    for (int i = t; i < total; i += TOPK_T) {
      float v = finals[i];
      if (v > bv || (v == bv && i < bi)) { bv = v; bi = i; }
    }
    wave_argmax(bv, bi);
    if (lane == 0) { rv[wid] = bv; ri[wid] = bi; }
    __syncthreads();
    if (t < 32) {
      float v = rv[lane];
      int   i = ri[lane];
      wave_argmax(v, i);
      if (lane == 0) {
        topv[k] = v;
        topi[k] = i;
        finals[i] = -3.0e30f;  // exclude from further passes
      }
    }
    __threadfence();
    __syncthreads();
  }
}

// ---------------- Kernel 5: gather preds + labels --------------------------
__global__ void nms_output_kernel(const float* __restrict__ bboxes,
                                  const float* __restrict__ topv,
                                  const int* __restrict__ topi,
                                  int m, int n, float* __restrict__ out) {
  int k = blockIdx.x * blockDim.x + threadIdx.x;
  if (k >= m) return;
  float v = topv[k];
  int idx = topi[k];
  int cls = idx / n;
  int b   = idx - cls * n;
  float* p = out + (size_t)k * 5;
  if (v > 0.0f) {
    p[0] = bboxes[b * 4 + 0];
    p[1] = bboxes[b * 4 + 1];
    p[2] = bboxes[b * 4 + 2];
    p[3] = bboxes[b * 4 + 3];
    p[4] = v;
    out[(size_t)5 * m + k] = (float)cls;
  } else {
    p[0] = p[1] = p[2] = p[3] = p[4] = 0.0f;
    out[(size_t)5 * m + k] = -1.0f;
  }
}

// ---------------------------------------------------------------------------
extern "C" void kernel_launch(void* const* d_in, const int* in_sizes, int n_in,
                              void* d_out, int out_size, void* d_ws, size_t ws_size,
                              hipStream_t stream) {
  const float* bboxes = (const float*)d_in[0];
  const float* scores = (const float*)d_in[1];
  const float* conf_p = (const float*)d_in[2];
  const float* nms_p  = (const float*)d_in[3];

  const int n    = in_sizes[0] / 4;        // 4000
  const int Ctot = in_sizes[1] / n;        // 81
  const int C    = Ctot - 1;               // 80 foreground classes
  const int m    = out_size / 6;           // 100 (5 pred cols + 1 label col)

  // workspace layout (floats)
  float* w       = (float*)d_ws;
  float* x1      = w + 0 * n;
  float* y1      = w + 1 * n;
  float* x2      = w + 2 * n;
  float* y2      = w + 3 * n;
  float* area    = w + 4 * n;
  float* scoresT = w + 5 * n;              // C*n floats
  float* finals  = scoresT;                // safe alias: each block reads its
                                           // own row into LDS before rewriting
  float* topv    = w + 5 * n + (size_t)C * n;
  int*   topi    = (int*)(topv + 128);

  nms_prep_kernel<<<(n + 255) / 256, 256, 0, stream>>>(bboxes, x1, y1, x2, y2, area, n);
  nms_transpose_kernel<<<(n * C + 255) / 256, 256, 0, stream>>>(scores, scoresT, n, C, Ctot);
  nms_class_kernel<<<C, NMS_T, 0, stream>>>(x1, y1, x2, y2, area, scoresT, finals,
                                            conf_p, nms_p, n);
  nms_topk_kernel<<<1, TOPK_T, 0, stream>>>(finals, n * C, m, topv, topi);
  nms_output_kernel<<<(m + 127) / 128, 128, 0, stream>>>(bboxes, topv, topi, m, n,
                                                         (float*)d_out);
}